// LSTM_30322469110082
// MI455X (gfx1250) — compile-verified
//
#include <hip/hip_runtime.h>
#include <math.h>

#define T_LEN   100
#define H_DIM   100
#define L_CNT   5
#define OUT_DIM 10
#define G4      (4 * H_DIM)      // 400 gate rows
#define NTHREADS 512
#define NWAVES  (NTHREADS / 32)
#define LDS_ROW 102              // pad: 102 mod 64 = 38 -> conflict-free float2 reads

typedef __attribute__((ext_vector_type(2))) float v2f;
typedef __attribute__((ext_vector_type(8))) float v8f;

__device__ __forceinline__ float sigmoidf_(float x) {
    return 1.0f / (1.0f + __expf(-x));
}

extern __shared__ float s_whh[];   // [400][102] dynamic LDS (163,200 B)

__global__ __launch_bounds__(NTHREADS)
void lstm_fused_kernel(const float* __restrict__ input,   // [100][100]
                       const float* __restrict__ Wih,     // [5][400][100]
                       const float* __restrict__ Whh,     // [5][400][100]
                       const float* __restrict__ bih,     // [5][400]
                       const float* __restrict__ bhh,     // [5][400]
                       const float* __restrict__ Wout,    // [10][100]
                       const float* __restrict__ bout,    // [10]
                       float* __restrict__ out,           // [100][10]
                       float* __restrict__ ws)
{
    __shared__ __align__(16) float s_h[H_DIM];
    __shared__ __align__(16) float s_c[H_DIM];
    __shared__ __align__(16) float s_g[G4];

    float* xg    = ws;                         // [100][400]
    float* xbuf0 = ws + (size_t)T_LEN * G4;    // [100][100]
    float* xbuf1 = xbuf0 + (size_t)T_LEN * H_DIM;

    const int tid  = threadIdx.x;
    const int lane = tid & 31;
    const int wid  = tid >> 5;
    const int m    = lane & 15;                // tile row (A) / tile col (B,C,D)
    const int hi   = lane >> 4;                // half-wave select
    const int koff = hi * 2;                   // K offset within 4-step per ISA layout

    for (int l = 0; l < L_CNT; ++l) {
        const float* xcur  = (l == 0) ? input : ((l & 1) ? xbuf0 : xbuf1);
        float*       xnext = (l & 1) ? xbuf1 : xbuf0;
        const float* wih_l = Wih + (size_t)l * G4 * H_DIM;
        const float* whh_l = Whh + (size_t)l * G4 * H_DIM;
        const float* bih_l = bih + l * G4;
        const float* bhh_l = bhh + l * G4;

        // ---- stage Whh[l] into LDS (padded rows), float2 vectorized ----
        {
            const v2f* src = (const v2f*)whh_l;             // coalesced b64 reads
            for (int i = tid; i < (G4 * H_DIM) / 2; i += NTHREADS) {
                int g  = i / (H_DIM / 2);
                int k2 = i - g * (H_DIM / 2);
                *(v2f*)(s_whh + g * LDS_ROW + 2 * k2) = src[i];  // 8B-aligned ds b64
            }
        }
        __threadfence_block();
        __syncthreads();

        // ---- XG = X * Wih^T + (bih+bhh) via v_wmma_f32_16x16x4_f32 ----
        const int rowTiles = (T_LEN + 15) / 16;   // 7
        const int colTiles = G4 / 16;             // 25
        for (int tile = wid; tile < rowTiles * colTiles; tile += NWAVES) {
            const int t0 = (tile / colTiles) * 16;
            const int g0 = (tile % colTiles) * 16;
            int trow = t0 + m;
            if (trow >= T_LEN) trow = T_LEN - 1;   // clamp; garbage rows never stored
            const float* arow = xcur  + (size_t)trow * H_DIM;
            const float* brow = wih_l + (size_t)(g0 + m) * H_DIM;  // B[k][n]=Wih[g0+n][k]

            v8f acc = {};
            for (int kk = 0; kk < H_DIM; kk += 4) {
                v2f a, b;
                a.x = arow[kk + koff];
                a.y = arow[kk + koff + 1];
                b.x = brow[kk + koff];
                b.y = brow[kk + koff + 1];
                acc = __builtin_amdgcn_wmma_f32_16x16x4_f32(
                        false, a, false, b, (short)0, acc, false, false);
            }

            const float biasv = bih_l[g0 + m] + bhh_l[g0 + m];
            const int rbase = t0 + (hi ? 8 : 0);   // D: VGPR j -> row j / j+8
            #pragma unroll
            for (int j = 0; j < 8; ++j) {
                int row = rbase + j;
                if (row < T_LEN)
                    xg[(size_t)row * G4 + g0 + m] = acc[j] + biasv;
            }
        }
        __threadfence_block();
        __syncthreads();

        // ---- recurrent scan (critical path) ----
        if (tid < H_DIM) { s_h[tid] = 0.0f; s_c[tid] = 0.0f; }
        __syncthreads();

        for (int t = 0; t < T_LEN; ++t) {
            if (tid < G4) {
                const v2f* wr = (const v2f*)(s_whh + tid * LDS_ROW); // conflict-free b64
                const v2f* hv = (const v2f*)s_h;                     // broadcast b64
                float acc = xg[(size_t)t * G4 + tid];
                #pragma unroll 5
                for (int k2 = 0; k2 < H_DIM / 2; ++k2) {
                    v2f w = wr[k2];
                    v2f h = hv[k2];
                    acc = fmaf(w.x, h.x, acc);
                    acc = fmaf(w.y, h.y, acc);
                }
                s_g[tid] = acc;
            }
            __syncthreads();
            if (tid < H_DIM) {
                float ig = sigmoidf_(s_g[tid]);
                float fg = sigmoidf_(s_g[tid + H_DIM]);
                float gg = tanhf(s_g[tid + 2 * H_DIM]);
                float og = sigmoidf_(s_g[tid + 3 * H_DIM]);
                float c  = fg * s_c[tid] + ig * gg;
                float h  = og * tanhf(c);
                s_c[tid] = c;
                s_h[tid] = h;
                xnext[(size_t)t * H_DIM + tid] = (l < L_CNT - 1) ? fmaxf(h, 0.0f) : h;
            }
            __syncthreads();
        }
        __threadfence_block();
        __syncthreads();
    }

    // ---- output projection via WMMA: out = sigmoid(X @ Wout^T + bout) ----
    // One 16-col tile (cols >= OUT_DIM zero-padded), 7 row tiles over waves.
    {
        const float* xf = xbuf0;   // layer 4 (even) wrote xbuf0
        const int rowTiles = (T_LEN + 15) / 16;   // 7
        for (int tile = wid; tile < rowTiles; tile += NWAVES) {
            const int t0 = tile * 16;
            int trow = t0 + m;
            if (trow >= T_LEN) trow = T_LEN - 1;
            const float* arow = xf + (size_t)trow * H_DIM;
            const bool colOk = (m < OUT_DIM);
            const float* brow = Wout + (size_t)(colOk ? m : 0) * H_DIM;

            v8f acc = {};
            for (int kk = 0; kk < H_DIM; kk += 4) {
                v2f a, b;
                a.x = arow[kk + koff];
                a.y = arow[kk + koff + 1];
                float bx = brow[kk + koff];
                float by = brow[kk + koff + 1];
                b.x = colOk ? bx : 0.0f;   // per-lane select keeps EXEC all-1s
                b.y = colOk ? by : 0.0f;
                acc = __builtin_amdgcn_wmma_f32_16x16x4_f32(
                        false, a, false, b, (short)0, acc, false, false);
            }

            const float biasv = colOk ? bout[m] : 0.0f;
            const int rbase = t0 + (hi ? 8 : 0);
            #pragma unroll
            for (int j = 0; j < 8; ++j) {
                int row = rbase + j;
                if (row < T_LEN && colOk)
                    out[(size_t)row * OUT_DIM + m] = sigmoidf_(acc[j] + biasv);
            }
        }
    }
}

extern "C" void kernel_launch(void* const* d_in, const int* in_sizes, int n_in,
                              void* d_out, int out_size, void* d_ws, size_t ws_size,
                              hipStream_t stream) {
    const float* input = (const float*)d_in[0];
    const float* Wih   = (const float*)d_in[1];
    const float* Whh   = (const float*)d_in[2];
    const float* bih   = (const float*)d_in[3];
    const float* bhh   = (const float*)d_in[4];
    const float* Wout  = (const float*)d_in[5];
    const float* bout  = (const float*)d_in[6];
    float* out = (float*)d_out;
    float* ws  = (float*)d_ws;

    const size_t shmem = (size_t)G4 * LDS_ROW * sizeof(float);  // 163,200 B
    (void)hipFuncSetAttribute((const void*)lstm_fused_kernel,
                              hipFuncAttributeMaxDynamicSharedMemorySize,
                              (int)shmem);

    lstm_fused_kernel<<<dim3(1), dim3(NTHREADS), shmem, stream>>>(
        input, Wih, Whh, bih, bhh, Wout, bout, out, ws);
}